// DynamicSkipConnection_16638703304904
// MI455X (gfx1250) — compile-verified
//
#include <hip/hip_runtime.h>
#include <hip/hip_bf16.h>

// ---------------------------------------------------------------------------
// DynamicSkipConnection: out[b,s,o] = sum_i x[b,s,i] * (w[o,i] * mask[o,i])
// mask has <= 100 nonzeros -> sparse scatter, bandwidth-bound (~240 MB total).
// Strategy: compact mask -> sorted (o,i,w) list; per-block LDS row accumulator;
// async global<->LDS data movement (gfx1250 ASYNCcnt path); streamed b128 out.
// ---------------------------------------------------------------------------

#define CAP 128                       // >= MAX_CONNECTIONS (100)

constexpr int IN_DIM    = 2048;
constexpr int OUT_DIM   = 2048;
constexpr int ROWS      = 16;         // (b,s) rows per block
constexpr int THREADS   = 256;        // 8 waves (wave32)
constexpr int ACC_STRIDE = OUT_DIM + 4;   // 2052 floats: rotate LDS banks per row
constexpr int CTR_STRIDE = CAP + 1;       // 129: conflict-free contrib reads

constexpr int ACC_BYTES  = ROWS * ACC_STRIDE * 4;   // 131,328
constexpr int CTR_BYTES  = ROWS * CTR_STRIDE * 4;   //   8,256
constexpr int META_BYTES = CAP * 12;                //   1,536
constexpr int SMEM_BYTES = ACC_BYTES + CTR_BYTES + META_BYTES;   // ~141 KB (< 320 KB/WGP)

// Pointer types matching the async builtins' signatures (from hipcc diagnostics):
//   load_b32 : (global int*,  shared int*,  imm, imm)
//   store_b128: (global v4i*, shared v4i*, imm, imm)
typedef int v4i __attribute__((ext_vector_type(4)));
typedef __attribute__((address_space(1))) int   g_i32;
typedef __attribute__((address_space(1))) v4i   g_v4i;
typedef __attribute__((address_space(3))) int   s_i32;
typedef __attribute__((address_space(3))) v4i   s_v4i;

// ---- gfx1250 async LDS<->global paths (guarded; fall back to plain vmem) ----
#if defined(__has_builtin)
#if __has_builtin(__builtin_amdgcn_global_store_async_from_lds_b128)
#define HAVE_ASYNC_STORE 1
#endif
#if __has_builtin(__builtin_amdgcn_global_load_async_to_lds_b32)
#define HAVE_ASYNC_LOAD 1
#endif
#if __has_builtin(__builtin_amdgcn_s_wait_asynccnt)
#define HAVE_WAIT_ASYNC_BUILTIN 1
#endif
#endif

__device__ __forceinline__ void wait_asynccnt0() {
#if defined(HAVE_WAIT_ASYNC_BUILTIN)
    __builtin_amdgcn_s_wait_asynccnt(0);
#else
    asm volatile("s_wait_asynccnt 0" ::: "memory");
#endif
}

// ---------------------------------------------------------------------------
// Kernel 1: reset the workspace counter (deterministic per launch).
// ---------------------------------------------------------------------------
__global__ void dsc_init_ws(int* __restrict__ cnt) {
    if (threadIdx.x == 0) *cnt = 0;
}

// ---------------------------------------------------------------------------
// Kernel 2: compact active (o,i) pairs + weights into d_ws.
// mask is a jnp bool array -> 1 byte per element.
// ---------------------------------------------------------------------------
__global__ void dsc_compact(const unsigned char* __restrict__ mask,
                            const float* __restrict__ w,
                            int* __restrict__ cnt,
                            int* __restrict__ o_idx,
                            int* __restrict__ i_idx,
                            float* __restrict__ wv,
                            int total) {
    int j = blockIdx.x * blockDim.x + threadIdx.x;
    if (j >= total) return;
    if (mask[j]) {
        int pos = atomicAdd(cnt, 1);
        if (pos < CAP) {
            o_idx[pos] = j / IN_DIM;
            i_idx[pos] = j % IN_DIM;
            wv[pos]    = w[j];
        }
    }
}

// ---------------------------------------------------------------------------
// Kernel 3: deterministic rank sort by (o,i).  Keys are unique (flat index),
// n <= 128 -> parallel O(n^2) work with O(n) depth; fixes FP accum order.
// ---------------------------------------------------------------------------
__global__ void dsc_sort(int* __restrict__ cnt,
                         int* __restrict__ o_idx,
                         int* __restrict__ i_idx,
                         float* __restrict__ wv) {
    __shared__ int   keys[CAP];
    __shared__ int   so[CAP], si[CAP];
    __shared__ float sw[CAP];
    int n = *cnt;
    if (n > CAP) n = CAP;
    int t = threadIdx.x;
    if (t < n) {
        so[t] = o_idx[t];
        si[t] = i_idx[t];
        sw[t] = wv[t];
        keys[t] = so[t] * IN_DIM + si[t];
    }
    __syncthreads();
    if (t < n) {
        int k = keys[t];
        int rank = 0;
        for (int a = 0; a < n; ++a) rank += (keys[a] < k) ? 1 : 0;
        o_idx[rank] = so[t];
        i_idx[rank] = si[t];
        wv[rank]    = sw[t];
    }
    if (t == 0) *cnt = n;
}

// ---------------------------------------------------------------------------
// Kernel 4: main compute.  Each block owns ROWS=16 output rows (8 KB each)
// in LDS.  Phases: zero acc -> async-gather x values into LDS -> deterministic
// per-row serial scatter -> async stream LDS rows to global (full 128B lines).
// ---------------------------------------------------------------------------
__global__ __launch_bounds__(THREADS)
void dsc_compute(const float* __restrict__ x,
                 const int* __restrict__ cnt,
                 const int* __restrict__ o_idx,
                 const int* __restrict__ i_idx,
                 const float* __restrict__ wv,
                 float* __restrict__ out) {
    extern __shared__ char smem[];
    float* acc     = (float*)smem;                             // ROWS*ACC_STRIDE
    float* contrib = (float*)(smem + ACC_BYTES);               // ROWS*CTR_STRIDE
    int*   s_o     = (int*)(smem + ACC_BYTES + CTR_BYTES);     // CAP
    int*   s_i     = s_o + CAP;                                // CAP
    float* s_w     = (float*)(s_i + CAP);                      // CAP
    __shared__ int s_n;

    const int t = threadIdx.x;
    const size_t rowBase = (size_t)blockIdx.x * ROWS;

    if (t == 0) s_n = min(*cnt, CAP);
    if (t < CAP) {
        s_o[t] = o_idx[t];
        s_i[t] = i_idx[t];
        s_w[t] = wv[t];
    }

    // Phase 1: zero the accumulator tile (ds_store_b128).
    {
        float4 z = make_float4(0.f, 0.f, 0.f, 0.f);
        float4* accv = (float4*)acc;
        const int nvec = ROWS * ACC_STRIDE / 4;   // ACC_STRIDE % 4 == 0
        for (int v = t; v < nvec; v += THREADS) accv[v] = z;
    }
    __syncthreads();
    const int n = s_n;

    // Phase 2: gather x[p, i_c] for all (row r, connection c).
    // 16 threads per c -> all 256 threads busy; x columns live in L2.
    {
        const int r = t & (ROWS - 1);
        for (int c = t >> 4; c < n; c += THREADS / ROWS) {
            const float* gp = x + (rowBase + (size_t)r) * IN_DIM + s_i[c];
            float* lp = contrib + r * CTR_STRIDE + c;
#if defined(HAVE_ASYNC_LOAD)
            __builtin_amdgcn_global_load_async_to_lds_b32(
                (g_i32*)gp, (s_i32*)lp, 0, 0);
#else
            *lp = *gp;
#endif
        }
    }
#if defined(HAVE_ASYNC_LOAD)
    wait_asynccnt0();
#endif
    __syncthreads();

    // Phase 3: deterministic scatter — one thread per row, fixed (sorted)
    // connection order, so duplicate-o accumulation order is reproducible.
    if (t < ROWS) {
        float* row = acc + t * ACC_STRIDE;          // bank-rotated per row
        const float* cr = contrib + t * CTR_STRIDE;
        for (int c = 0; c < n; ++c) row[s_o[c]] += cr[c] * s_w[c];
    }
    __syncthreads();

    // Phase 4: stream the 16 rows to global memory. Block output region is
    // contiguous per row; b128 stores saturate HBM write bandwidth.
    {
        const int VPR = OUT_DIM / 4;                 // 512 float4 per row
        for (int v = t; v < ROWS * VPR; v += THREADS) {
            int r  = v >> 9;                         // v / VPR
            int c4 = v & (VPR - 1);
            float* gp = out + (rowBase + (size_t)r) * OUT_DIM + 4 * c4;
            float* lp = acc + r * ACC_STRIDE + 4 * c4;
#if defined(HAVE_ASYNC_STORE)
            __builtin_amdgcn_global_store_async_from_lds_b128(
                (g_v4i*)gp, (s_v4i*)lp, 0, 0);
#else
            *(float4*)gp = *(const float4*)lp;
#endif
        }
    }
#if defined(HAVE_ASYNC_STORE)
    wait_asynccnt0();
#endif
}

// ---------------------------------------------------------------------------
// Host launcher.  Inputs: x [4,4096,2048] f32, skip_weight [2048,2048] f32,
// active_mask [2048,2048] bool (1 byte/elem).  Output: [4,4096,2048] f32.
// ---------------------------------------------------------------------------
extern "C" void kernel_launch(void* const* d_in, const int* in_sizes, int n_in,
                              void* d_out, int out_size, void* d_ws, size_t ws_size,
                              hipStream_t stream) {
    const float*         x    = (const float*)d_in[0];
    const float*         w    = (const float*)d_in[1];
    const unsigned char* mask = (const unsigned char*)d_in[2];
    float*               out  = (float*)d_out;

    // Workspace layout: [count pad pad pad][o_idx[CAP]][i_idx[CAP]][wv[CAP]]
    int*   cnt   = (int*)d_ws;
    int*   o_idx = (int*)((char*)d_ws + 16);
    int*   i_idx = o_idx + CAP;
    float* wv    = (float*)(i_idx + CAP);

    const int total = in_sizes[1];               // OUT_DIM * IN_DIM
    dsc_init_ws<<<1, 32, 0, stream>>>(cnt);
    dsc_compact<<<(total + THREADS - 1) / THREADS, THREADS, 0, stream>>>(
        mask, w, cnt, o_idx, i_idx, wv, total);
    dsc_sort<<<1, CAP, 0, stream>>>(cnt, o_idx, i_idx, wv);

    const int rows   = out_size / OUT_DIM;       // B*S = 16384
    const int blocks = rows / ROWS;              // 1024
    dsc_compute<<<blocks, THREADS, SMEM_BYTES, stream>>>(
        x, cnt, o_idx, i_idx, wv, out);
}